// SheafLayer_48893907698074
// MI455X (gfx1250) — compile-verified
//
#include <hip/hip_runtime.h>
#include <hip/hip_bf16.h>

typedef __attribute__((ext_vector_type(2))) float v2f;
typedef __attribute__((ext_vector_type(8))) float v8f;

#define EPG      16      // edges per workgroup
#define PAD      33      // LDS row pitch (floats) -> bank-conflict-free
#define NTHREADS 128     // 4 waves (wave32)

__device__ __forceinline__ v8f wmma_f32_16x16x4(v2f a, v2f b, v8f c) {
  // D = A(16x4,f32) * B(4x16,f32) + C(16x16,f32)
  return __builtin_amdgcn_wmma_f32_16x16x4_f32(
      /*neg_a=*/false, a, /*neg_b=*/false, b,
      /*c_mod=*/(short)0, c, /*reuse_a=*/false, /*reuse_b=*/false);
}

// ASYNCcnt-tracked global->LDS copy, 4B per lane, no VGPR round-trip.
__device__ __forceinline__ void async_cp_b32(const float* gsrc, const float* ldst) {
  unsigned loff = (unsigned)(uintptr_t)ldst;   // low 32 bits = LDS offset
  asm volatile("global_load_async_to_lds_b32 %0, %1, off"
               :: "v"(loff), "v"(gsrc) : "memory");
}

// Issue 16 async copies per thread: two 32x32 tiles into padded LDS buffers.
__device__ __forceinline__ void stage_x_async(const float* __restrict__ xs,
                                              const float* __restrict__ xd,
                                              float* xlb, float* xrb, int tid) {
#pragma unroll
  for (int i = tid; i < 1024; i += NTHREADS) {   // 8 iterations
    int r = i >> 5, cc = i & 31;
    async_cp_b32(xs + i, xlb + r * PAD + cc);
    async_cp_b32(xd + i, xrb + r * PAD + cc);
  }
}

__global__ void zero_out_kernel(float* __restrict__ out, int n) {
  int i = blockIdx.x * blockDim.x + threadIdx.x;
  if (i < n) out[i] = 0.0f;
}

__global__ __launch_bounds__(NTHREADS) void sheaf_edge_kernel(
    const float* __restrict__ x,     // [n_nodes][32][32]
    const int*   __restrict__ eidx,  // [2][n_edges]
    const float* __restrict__ attr,  // [n_edges][32]
    const float* __restrict__ rmap,  // [32][2048]  (= restriction_maps reshaped)
    float* __restrict__ out,         // [n_nodes][32][32]
    int n_edges)
{
  // F for 16 edges, both sides, padded: [EPG][2][32][PAD]
  __shared__ float F_lds[EPG * 2 * 32 * PAD];
  __shared__ float attr_lds[EPG * PAD];
  __shared__ float xl_lds[2][32 * PAD];   // double-buffered async staging
  __shared__ float xr_lds[2][32 * PAD];
  __shared__ float dx_lds[32 * PAD];

  const int tid  = threadIdx.x;
  const int lane = tid & 31;
  const int wid  = tid >> 5;
  const int l16  = lane & 15;
  const int half = lane >> 4;      // 0: lanes 0-15, 1: lanes 16-31
  const int ti   = wid >> 1;       // this wave's output tile row (0/1)
  const int tj   = wid & 1;        // this wave's output tile col (0/1)

  const int e0 = blockIdx.x * EPG;
  const int elast = n_edges - 1;

  // ---- prologue: kick off edge-0 x-tile copies (overlaps whole F-phase) ----
  {
    int ge = (e0 < n_edges) ? e0 : elast;
    int src = eidx[ge], dst = eidx[n_edges + ge];
    stage_x_async(x + (size_t)src * 1024, x + (size_t)dst * 1024,
                  xl_lds[0], xr_lds[0], tid);
  }

  // ---- stage edge_attr tile (16 edges x 32 attrs) ----
  for (int i = tid; i < EPG * 32; i += NTHREADS) {
    int e = i >> 5, a = i & 31;
    int ge = e0 + e;
    attr_lds[e * PAD + a] = (ge < n_edges) ? attr[ge * 32 + a] : 0.0f;
  }
  __syncthreads();

  // ---- Phase 1: F(16x2048) = attr(16x32) @ R(32x2048); wave w owns 512 cols ----
  {
    v2f afrag[8];
#pragma unroll
    for (int s = 0; s < 8; ++s) {              // A frag: A[m][k], m=l16, k=s*4+half*2+v
      int k = s * 4 + half * 2;
      afrag[s].x = attr_lds[l16 * PAD + k];
      afrag[s].y = attr_lds[l16 * PAD + k + 1];
    }
    for (int t = 0; t < 32; ++t) {             // 32 tiles of 16 cols per wave
      int colbase = wid * 512 + t * 16;
      v8f c = {};
#pragma unroll
      for (int s = 0; s < 8; ++s) {            // K = 32 in 8 steps of 4
        int k = s * 4 + half * 2;
        v2f b;                                 // B frag: B[k][n], n=l16
        b.x = rmap[(k)     * 2048 + colbase + l16];
        b.y = rmap[(k + 1) * 2048 + colbase + l16];
        c = wmma_f32_16x16x4(afrag[s], b, c);
      }
      // D layout: VGPR v -> (M = v + half*8, N = l16); M = edge-in-group
      int col  = colbase + l16;                // 0..2047
      int side = col >> 10;                    // 0 = left, 1 = right
      int row  = (col & 1023) >> 5;            // i of F[i][j]
      int jj   = col & 31;                     // j
      float* Fb = &F_lds[(side * 32 + row) * PAD + jj];
#pragma unroll
      for (int v = 0; v < 8; ++v)
        Fb[(v + half * 8) * (2 * 32 * PAD)] = c[v];
    }
  }

  // ---- Phase 2: per-edge dx and transposed scatter products ----
  for (int e = 0; e < EPG; ++e) {
    int ge = e0 + e;
    if (ge >= n_edges) break;                  // uniform across block
    int src = eidx[ge];
    int dst = eidx[n_edges + ge];
    const int cur = e & 1;

    // prefetch edge e+1 into the other buffer (clamped: always 16 asyncs/wave)
    {
      int gn = e0 + e + 1;
      if (gn > elast) gn = elast;
      int nsrc = eidx[gn], ndst = eidx[n_edges + gn];
      stage_x_async(x + (size_t)nsrc * 1024, x + (size_t)ndst * 1024,
                    xl_lds[cur ^ 1], xr_lds[cur ^ 1], tid);
    }
    // ASYNCcnt completes in order: <=16 outstanding => edge e's 16 are done.
    asm volatile("s_wait_asynccnt 0x10" ::: "memory");
    __syncthreads();                            // make all waves' copies visible

    const float* Fl = &F_lds[(e * 2 + 0) * 32 * PAD];
    const float* Fr = &F_lds[(e * 2 + 1) * 32 * PAD];
    const float* xl = xl_lds[cur];
    const float* xr = xr_lds[cur];

    // dx tile (ti,tj) = sum_k Fl[ti*16+m][k]*xl[k][tj*16+n] - Fr.. * xr..
    v8f cdx = {};
#pragma unroll
    for (int s = 0; s < 8; ++s) {
      int k = s * 4 + half * 2;
      v2f a, b;
      a.x = Fl[(ti * 16 + l16) * PAD + k];
      a.y = Fl[(ti * 16 + l16) * PAD + k + 1];
      b.x = xl[(k)     * PAD + tj * 16 + l16];
      b.y = xl[(k + 1) * PAD + tj * 16 + l16];
      cdx = wmma_f32_16x16x4(a, b, cdx);
    }
#pragma unroll
    for (int s = 0; s < 8; ++s) {
      int k = s * 4 + half * 2;
      v2f a, b;
      a.x = -Fr[(ti * 16 + l16) * PAD + k];    // negate A => subtract
      a.y = -Fr[(ti * 16 + l16) * PAD + k + 1];
      b.x = xr[(k)     * PAD + tj * 16 + l16];
      b.y = xr[(k + 1) * PAD + tj * 16 + l16];
      cdx = wmma_f32_16x16x4(a, b, cdx);
    }
#pragma unroll
    for (int v = 0; v < 8; ++v)
      dx_lds[(ti * 16 + half * 8 + v) * PAD + tj * 16 + l16] = cdx[v];
    __syncthreads();

    // xe tiles: A = F^T  (A[m][k] = F[k][ti*16+m], just index-swapped LDS reads)
    v8f cl = {}, cr = {};
#pragma unroll
    for (int s = 0; s < 8; ++s) {
      int k = s * 4 + half * 2;
      v2f b, al, ar;
      b.x  = dx_lds[(k)     * PAD + tj * 16 + l16];
      b.y  = dx_lds[(k + 1) * PAD + tj * 16 + l16];
      al.x = Fl[(k)     * PAD + ti * 16 + l16];
      al.y = Fl[(k + 1) * PAD + ti * 16 + l16];
      ar.x = Fr[(k)     * PAD + ti * 16 + l16];
      ar.y = Fr[(k + 1) * PAD + ti * 16 + l16];
      cl = wmma_f32_16x16x4(al, b, cl);
      cr = wmma_f32_16x16x4(ar, b, cr);
    }

    // scatter: out[src] += cl tile; out[dst] -= cr tile (coalesced f32 atomics)
    float* po = out + (size_t)src * 1024 + (ti * 16) * 32 + tj * 16;
    float* qo = out + (size_t)dst * 1024 + (ti * 16) * 32 + tj * 16;
#pragma unroll
    for (int v = 0; v < 8; ++v) {
      int m = half * 8 + v;
      atomicAdd(po + m * 32 + l16,  cl[v]);
      atomicAdd(qo + m * 32 + l16, -cr[v]);
    }
    // next iteration's post-wait barrier orders dx_lds / x-buffer reuse
  }
  // leftover prefetch asyncs are drained by s_endpgm's implicit wait-idle
}

extern "C" void kernel_launch(void* const* d_in, const int* in_sizes, int n_in,
                              void* d_out, int out_size, void* d_ws, size_t ws_size,
                              hipStream_t stream) {
  const float* x    = (const float*)d_in[0];
  const int*   eidx = (const int*)d_in[1];   // edge_index [2][E]
  const float* attr = (const float*)d_in[2]; // [E][32]
  const float* rmap = (const float*)d_in[3]; // [32][2][32][32] == [32][2048]
  float* out = (float*)d_out;

  const int n_edges = in_sizes[2] / 32;

  // out is poisoned by the harness: zero it every call (accumulated via atomics)
  zero_out_kernel<<<(out_size + 255) / 256, 256, 0, stream>>>(out, out_size);

  const int nblk = (n_edges + EPG - 1) / EPG;
  sheaf_edge_kernel<<<nblk, NTHREADS, 0, stream>>>(x, eidx, attr, rmap, out, n_edges);
}